// LinearSelfAttention_13477607375014
// MI455X (gfx1250) — compile-verified
//
#include <hip/hip_runtime.h>
#include <math.h>

#define DIM     256
#define HALF_D  128
#define NTOK    8192
#define NB      4
#define ROWS    (NB * NTOK)     // 32768
#define EPSN    1e-5f
#define SPLIT   8

typedef float v2f __attribute__((ext_vector_type(2)));
typedef float v8f __attribute__((ext_vector_type(8)));

__device__ __forceinline__ v8f wmma_f32(v2f a, v2f b, v8f c) {
  // D = A(16x4 f32) x B(4x16 f32) + C(16x16 f32)
  return __builtin_amdgcn_wmma_f32_16x16x4_f32(
      /*neg_a=*/false, a, /*neg_b=*/false, b,
      /*c_mod=*/(short)0, c, /*reuse_a=*/false, /*reuse_b=*/false);
}

// ---------------------------------------------------------------------------
__global__ __launch_bounds__(256)
void lsa_zero_kernel(float* __restrict__ p, int n) {
  int i = blockIdx.x * blockDim.x + threadIdx.x;
  if (i < n) p[i] = 0.0f;
}

// ---------------------------------------------------------------------------
// Fused q/k/v GEMM + q-rotary + raw k/v store + weighted norm accumulation.
// grid.x = ROWS/16 row tiles; 8 waves/block; wave w owns col pair (16w, 16w+128).
__global__ __launch_bounds__(256)
void lsa_qkv_kernel(const float* __restrict__ phi, const float* __restrict__ coords,
                    const float* __restrict__ weights,
                    const float* __restrict__ Wq, const float* __restrict__ bq,
                    const float* __restrict__ Wk, const float* __restrict__ bk,
                    const float* __restrict__ Wv, const float* __restrict__ bv,
                    const float* __restrict__ lift,
                    float* __restrict__ qbuf, float* __restrict__ kbuf,
                    float* __restrict__ vbuf,
                    float* __restrict__ normk, float* __restrict__ normv)
{
  const int lane = threadIdx.x & 31;
  const int w    = threadIdx.x >> 5;          // 0..7
  const int r0   = blockIdx.x * 16;           // global row base
  const int b    = r0 / NTOK;
  const int m    = lane & 15;                 // A row / BCD col (n)
  const int half = lane >> 4;
  const int j    = w * 16 + m;                // column in [0,128)

  v8f aq0{}, aq1{}, ak0{}, ak1{}, av0{}, av1{};

  const float* phiRow = phi + (size_t)(r0 + m) * DIM;
  const float* wq0 = Wq + (size_t)j * DIM;          // B[k][n] = W[col][k]
  const float* wq1 = Wq + (size_t)(j + 128) * DIM;
  const float* wk0 = Wk + (size_t)j * DIM;
  const float* wk1 = Wk + (size_t)(j + 128) * DIM;
  const float* wv0 = Wv + (size_t)j * DIM;
  const float* wv1 = Wv + (size_t)(j + 128) * DIM;

  for (int kk = 0; kk < DIM; kk += 4) {
    const int ko = kk + 2 * half;
    v2f a   = *(const v2f*)(phiRow + ko);
    v2f bq0 = *(const v2f*)(wq0 + ko);
    v2f bq1 = *(const v2f*)(wq1 + ko);
    v2f bk0 = *(const v2f*)(wk0 + ko);
    v2f bk1 = *(const v2f*)(wk1 + ko);
    v2f bv0 = *(const v2f*)(wv0 + ko);
    v2f bv1 = *(const v2f*)(wv1 + ko);
    aq0 = wmma_f32(a, bq0, aq0);
    aq1 = wmma_f32(a, bq1, aq1);
    ak0 = wmma_f32(a, bk0, ak0);
    ak1 = wmma_f32(a, bk1, ak1);
    av0 = wmma_f32(a, bv0, av0);
    av1 = wmma_f32(a, bv1, av1);
  }

  const float bqv1 = bq[j], bqv2 = bq[j + 128];
  const float bkv1 = bk[j], bkv2 = bk[j + 128];
  const float bvv1 = bv[j], bvv2 = bv[j + 128];
  const float lx = lift[j * 3 + 0], ly = lift[j * 3 + 1], lz = lift[j * 3 + 2];

  float pk0 = 0.f, pk1 = 0.f, pv0 = 0.f, pv1 = 0.f;
  #pragma unroll
  for (int i = 0; i < 8; ++i) {
    const int row = r0 + i + 8 * half;
    // --- q: bias + rotary, store rotated
    float f1 = aq0[i] + bqv1;
    float f2 = aq1[i] + bqv2;
    float cx = coords[(size_t)row * 3 + 0];
    float cy = coords[(size_t)row * 3 + 1];
    float cz = coords[(size_t)row * 3 + 2];
    float ph = cx * lx + cy * ly + cz * lz;
    float s, c;
    __sincosf(ph, &s, &c);
    qbuf[(size_t)row * DIM + j]       = f1 * c - f2 * s;
    qbuf[(size_t)row * DIM + j + 128] = f1 * s + f2 * c;
    // --- k/v: bias, raw store, weighted norm partials
    const float wt = weights[row];
    float k1 = ak0[i] + bkv1;
    float k2 = ak1[i] + bkv2;
    kbuf[(size_t)row * DIM + j]       = k1;
    kbuf[(size_t)row * DIM + j + 128] = k2;
    pk0 += wt * k1 * k1;
    pk1 += wt * k2 * k2;
    float x1 = av0[i] + bvv1;
    float x2 = av1[i] + bvv2;
    vbuf[(size_t)row * DIM + j]       = x1;
    vbuf[(size_t)row * DIM + j + 128] = x2;
    pv0 += wt * x1 * x1;
    pv1 += wt * x2 * x2;
  }
  unsafeAtomicAdd(&normk[b * DIM + j],       pk0);
  unsafeAtomicAdd(&normk[b * DIM + j + 128], pk1);
  unsafeAtomicAdd(&normv[b * DIM + j],       pv0);
  unsafeAtomicAdd(&normv[b * DIM + j + 128], pv1);
}

// ---------------------------------------------------------------------------
// Finalize: k <- weights * rotary(k * rsqrt(normk+eps)); v <- v * rsqrt(normv+eps)
__global__ __launch_bounds__(256)
void lsa_normrot_kernel(const float* __restrict__ coords,
                        const float* __restrict__ weights,
                        const float* __restrict__ lift,
                        const float* __restrict__ normk,
                        const float* __restrict__ normv,
                        float* __restrict__ kbuf, float* __restrict__ vbuf)
{
  int t = blockIdx.x * blockDim.x + threadIdx.x;
  if (t >= ROWS * HALF_D) return;
  const int j   = t & (HALF_D - 1);
  const int row = t >> 7;
  const int b   = row / NTOK;

  const float ik1 = rsqrtf(normk[b * DIM + j] + EPSN);
  const float ik2 = rsqrtf(normk[b * DIM + j + 128] + EPSN);
  const float iv1 = rsqrtf(normv[b * DIM + j] + EPSN);
  const float iv2 = rsqrtf(normv[b * DIM + j + 128] + EPSN);

  float k1 = kbuf[(size_t)row * DIM + j] * ik1;
  float k2 = kbuf[(size_t)row * DIM + j + 128] * ik2;
  float ph = coords[(size_t)row * 3 + 0] * lift[j * 3 + 0] +
             coords[(size_t)row * 3 + 1] * lift[j * 3 + 1] +
             coords[(size_t)row * 3 + 2] * lift[j * 3 + 2];
  float s, c;
  __sincosf(ph, &s, &c);
  const float wt = weights[row];
  kbuf[(size_t)row * DIM + j]       = wt * (k1 * c - k2 * s);
  kbuf[(size_t)row * DIM + j + 128] = wt * (k1 * s + k2 * c);
  vbuf[(size_t)row * DIM + j]       *= iv1;
  vbuf[(size_t)row * DIM + j + 128] *= iv2;
}

// ---------------------------------------------------------------------------
// kv[b] = k_fin^T @ v_fin  (reduce over tokens, split-N + atomic reduce)
__global__ __launch_bounds__(32)
void lsa_kv_kernel(const float* __restrict__ kbuf, const float* __restrict__ vbuf,
                   float* __restrict__ kv)
{
  const int lane = threadIdx.x & 31;
  int id = blockIdx.x;
  const int split = id % SPLIT; id /= SPLIT;
  const int et = id & 15; id >>= 4;
  const int dt = id & 15; id >>= 4;
  const int b  = id;
  const int m = lane & 15, half = lane >> 4;
  const int d0 = dt * 16, e0 = et * 16;

  const float* kb = kbuf + (size_t)b * NTOK * DIM;
  const float* vb = vbuf + (size_t)b * NTOK * DIM;
  const int chunk = NTOK / SPLIT;
  const int tbeg = split * chunk, tend = tbeg + chunk;

  v8f acc{};
  for (int tok = tbeg; tok < tend; tok += 4) {
    const int t0 = tok + 2 * half;
    v2f a, bb;
    a.x  = kb[(size_t)t0 * DIM + d0 + m];        // A[m][k] = k[tok][d0+m]
    a.y  = kb[(size_t)(t0 + 1) * DIM + d0 + m];
    bb.x = vb[(size_t)t0 * DIM + e0 + m];        // B[k][n] = v[tok][e0+n]
    bb.y = vb[(size_t)(t0 + 1) * DIM + e0 + m];
    acc = wmma_f32(a, bb, acc);
  }
  float* out = kv + (size_t)b * DIM * DIM;
  #pragma unroll
  for (int i = 0; i < 8; ++i)
    unsafeAtomicAdd(&out[(size_t)(d0 + i + 8 * half) * DIM + e0 + m], acc[i]);
}

// ---------------------------------------------------------------------------
// M[b] = kv[b] @ Wo^T
__global__ __launch_bounds__(32)
void lsa_mo_kernel(const float* __restrict__ kv, const float* __restrict__ Wo,
                   float* __restrict__ Mo)
{
  const int lane = threadIdx.x & 31;
  int id = blockIdx.x;
  const int et = id & 15; id >>= 4;
  const int dt = id & 15; id >>= 4;
  const int b  = id;
  const int m = lane & 15, half = lane >> 4;
  const int d0 = dt * 16, e0 = et * 16;

  const float* arow = kv + (size_t)b * DIM * DIM + (size_t)(d0 + m) * DIM;
  const float* brow = Wo + (size_t)(e0 + m) * DIM;   // B[k][n] = Wo[e0+n][k]

  v8f acc{};
  for (int f = 0; f < DIM; f += 4) {
    const int fo = f + 2 * half;
    v2f a  = *(const v2f*)(arow + fo);
    v2f bb = *(const v2f*)(brow + fo);
    acc = wmma_f32(a, bb, acc);
  }
  float* out = Mo + (size_t)b * DIM * DIM;
  #pragma unroll
  for (int i = 0; i < 8; ++i)
    out[(size_t)(d0 + i + 8 * half) * DIM + e0 + m] = acc[i];
}

// ---------------------------------------------------------------------------
// out = q_rot @ M[b] + bo
__global__ __launch_bounds__(256)
void lsa_out_kernel(const float* __restrict__ qbuf, const float* __restrict__ Mo,
                    const float* __restrict__ bo, float* __restrict__ out)
{
  const int lane = threadIdx.x & 31;
  const int w    = threadIdx.x >> 5;
  const int r0   = blockIdx.x * 16;
  const int b    = r0 / NTOK;
  const int m    = lane & 15, half = lane >> 4;
  const int e0a  = w * 16, e0b = w * 16 + 128;

  const float* qRow = qbuf + (size_t)(r0 + m) * DIM;
  const float* Mb   = Mo + (size_t)b * DIM * DIM;

  v8f acc0{}, acc1{};
  for (int kk = 0; kk < DIM; kk += 4) {
    const int ko = kk + 2 * half;
    v2f a = *(const v2f*)(qRow + ko);
    v2f b0, b1;
    b0.x = Mb[(size_t)ko * DIM + e0a + m];
    b0.y = Mb[(size_t)(ko + 1) * DIM + e0a + m];
    b1.x = Mb[(size_t)ko * DIM + e0b + m];
    b1.y = Mb[(size_t)(ko + 1) * DIM + e0b + m];
    acc0 = wmma_f32(a, b0, acc0);
    acc1 = wmma_f32(a, b1, acc1);
  }
  const float bo0 = bo[e0a + m], bo1 = bo[e0b + m];
  #pragma unroll
  for (int i = 0; i < 8; ++i) {
    const int row = r0 + i + 8 * half;
    out[(size_t)row * DIM + e0a + m] = acc0[i] + bo0;
    out[(size_t)row * DIM + e0b + m] = acc1[i] + bo1;
  }
}

// ---------------------------------------------------------------------------
extern "C" void kernel_launch(void* const* d_in, const int* in_sizes, int n_in,
                              void* d_out, int out_size, void* d_ws, size_t ws_size,
                              hipStream_t stream) {
  const float* phi     = (const float*)d_in[0];
  const float* coords  = (const float*)d_in[1];
  const float* weights = (const float*)d_in[2];
  const float* Wq      = (const float*)d_in[3];
  const float* bq      = (const float*)d_in[4];
  const float* Wk      = (const float*)d_in[5];
  const float* bk      = (const float*)d_in[6];
  const float* Wv      = (const float*)d_in[7];
  const float* bv      = (const float*)d_in[8];
  const float* Wo      = (const float*)d_in[9];
  const float* bo      = (const float*)d_in[10];
  const float* lift    = (const float*)d_in[11];
  float* out = (float*)d_out;

  // workspace layout (floats)
  const size_t Q = (size_t)ROWS * DIM;       // 8,388,608
  float* ws    = (float*)d_ws;
  float* qbuf  = ws;
  float* kbuf  = ws + Q;
  float* vbuf  = ws + 2 * Q;
  float* kv    = ws + 3 * Q;                 // NB*DIM*DIM = 262144
  float* normk = kv + (size_t)NB * DIM * DIM;    // 1024
  float* normv = normk + NB * DIM;               // 1024
  float* Mo    = normv + NB * DIM;               // 262144

  // zero atomically-accumulated buffers (kv | normk | normv are contiguous)
  const int nz = NB * DIM * DIM + 2 * NB * DIM;  // 264192
  lsa_zero_kernel<<<(nz + 255) / 256, 256, 0, stream>>>(kv, nz);

  // fused q/k/v GEMM + rotary(q) + norm accumulation
  lsa_qkv_kernel<<<ROWS / 16, 256, 0, stream>>>(
      phi, coords, weights, Wq, bq, Wk, bk, Wv, bv, lift,
      qbuf, kbuf, vbuf, normk, normv);

  // normalize + rotary(k) + weight; normalize v
  lsa_normrot_kernel<<<(ROWS * HALF_D + 255) / 256, 256, 0, stream>>>(
      coords, weights, lift, normk, normv, kbuf, vbuf);

  // kv = k^T v (split-N with atomic reduce)
  lsa_kv_kernel<<<NB * 16 * 16 * SPLIT, 32, 0, stream>>>(kbuf, vbuf, kv);

  // M = kv @ Wo^T
  lsa_mo_kernel<<<NB * 16 * 16, 32, 0, stream>>>(kv, Wo, Mo);

  // out = q_rot @ M + bo
  lsa_out_kernel<<<ROWS / 16, 256, 0, stream>>>(qbuf, Mo, bo, out);
}